// Brick_Wall_88759794139696
// MI455X (gfx1250) — compile-verified
//
#include <hip/hip_runtime.h>

// CDNA5 / gfx1250: wave32, WMMA f32 16x16x4.
typedef __attribute__((ext_vector_type(2))) float v2f;
typedef __attribute__((ext_vector_type(8))) float v8f;

#define WAVES_PER_BLOCK 4
#define NQ      4096     // num_qubits
#define SROW    8192     // covariance/update row stride
#define NTERMS  8        // Taylor terms
#define NSQUARE 6        // squarings (scale = 2^-6)

// E[a][b] of the 4x4 antisymmetric Majorana-coefficient matrix (x4 factor folded in).
__device__ __forceinline__ float eent(int a, int b,
                                      float c0, float c1, float c2,
                                      float c3, float c4, float c5) {
  if (a == b) return 0.0f;
  const int lo = a < b ? a : b;
  const int hi = a < b ? b : a;
  float v;
  if      (lo == 0 && hi == 1) v = -2.0f * c4;
  else if (lo == 0 && hi == 2) v =  2.0f * c2;
  else if (lo == 0 && hi == 3) v =  2.0f * c3;
  else if (lo == 1 && hi == 2) v = -2.0f * c0;
  else if (lo == 1 && hi == 3) v = -2.0f * c1;
  else                         v = -2.0f * c5;  // (2,3)
  return (a < b) ? v : -v;
}

// D_j = 4*PHC[j] : constant +-2 pattern.
__device__ __forceinline__ float dent(int j, int r, int c) {
  float v = 0.0f;
  switch (j) {
    case 0:  if (r==1 && c==2) v = -2.0f; else if (r==2 && c==1) v =  2.0f; break;
    case 1:  if (r==1 && c==3) v = -2.0f; else if (r==3 && c==1) v =  2.0f; break;
    case 2:  if (r==0 && c==2) v =  2.0f; else if (r==2 && c==0) v = -2.0f; break;
    case 3:  if (r==0 && c==3) v =  2.0f; else if (r==3 && c==0) v = -2.0f; break;
    case 4:  if (r==0 && c==1) v = -2.0f; else if (r==1 && c==0) v =  2.0f; break;
    default: if (r==2 && c==3) v = -2.0f; else if (r==3 && c==2) v =  2.0f; break;
  }
  return v;
}

// One 16x16 = 16x16 @ 16x16 matmul via 4x v_wmma_f32_16x16x4_f32.
// A fragments pre-loaded; B fragments pulled from the wave-private LDS tile.
__device__ __forceinline__ v8f mm16(const v2f A[4], const float* tile,
                                    int half, int l) {
  v8f acc = {0.f,0.f,0.f,0.f,0.f,0.f,0.f,0.f};
#pragma unroll
  for (int q = 0; q < 4; ++q) {
    v2f b;
    b.x = tile[(4*q + 2*half + 0) * 16 + l];
    b.y = tile[(4*q + 2*half + 1) * 16 + l];
    acc = __builtin_amdgcn_wmma_f32_16x16x4_f32(
        /*neg_a=*/false, A[q], /*neg_b=*/false, b,
        /*c_mod=*/(short)0, acc, /*reuse_a=*/false, /*reuse_b=*/false);
  }
  return acc;
}

__global__ void __launch_bounds__(WAVES_PER_BLOCK * 32)
brickwall_expm_kernel(const float* __restrict__ pcpa,
                      const float* __restrict__ cov,
                      const float* __restrict__ upd,
                      const float* __restrict__ chi,
                      const int*   __restrict__ layer,
                      float*       __restrict__ out) {
  __shared__ float tiles[WAVES_PER_BLOCK][16 * 16];
  const int lane = threadIdx.x & 31;
  const int w    = threadIdx.x >> 5;
  float* tile = tiles[w];
  const int wid = blockIdx.x * WAVES_PER_BLOCK + w;

  const int even  = ((layer[0] & 1) == 0) ? 1 : 0;
  const int ng    = even ? (NQ / 2) : (NQ / 2 - 1);
  const int boff  = even ? 0 : 2;
  const int nprob = 6 * ng;

  const int p0 = 2 * wid;
  if (p0 >= nprob) return;              // wave-uniform exit
  const bool has1 = (p0 + 1 < nprob);
  const int j0 = p0 / ng, i0 = p0 - j0 * ng;
  const int p1 = has1 ? (p0 + 1) : p0;
  const int j1 = p1 / ng, i1 = p1 - j1 * ng;

  // chi for both problems (broadcast loads, L2-resident)
  const float a0 = chi[i0*6+0], a1 = chi[i0*6+1], a2 = chi[i0*6+2],
              a3 = chi[i0*6+3], a4 = chi[i0*6+4], a5 = chi[i0*6+5];
  const float b0 = chi[i1*6+0], b1 = chi[i1*6+1], b2 = chi[i1*6+2],
              b3 = chi[i1*6+3], b4 = chi[i1*6+4], b5 = chi[i1*6+5];

  // Build X = diag(M0, M1) / 2^NSQUARE into the wave-private LDS tile.
  const float invs = 1.0f / 64.0f;
#pragma unroll
  for (int t = 0; t < 8; ++t) {
    const int linear = lane + 32 * t;        // 0..255 across the wave
    const int m = linear >> 4, n = linear & 15;
    float val = 0.0f;
    if ((m < 8) == (n < 8)) {                // on-diagonal 8x8 block
      const int mm = m & 7, nn = n & 7;
      const bool hi = (m >= 8);
      const float e0 = hi ? b0 : a0, e1 = hi ? b1 : a1, e2 = hi ? b2 : a2,
                  e3 = hi ? b3 : a3, e4 = hi ? b4 : a4, e5 = hi ? b5 : a5;
      const int jj = hi ? j1 : j0;
      if (mm < 4 && nn < 4)       val = eent(mm, nn, e0,e1,e2,e3,e4,e5);
      else if (mm < 4)            val = dent(jj, mm, nn - 4);
      else if (nn >= 4)           val = eent(mm - 4, nn - 4, e0,e1,e2,e3,e4,e5);
      // lower-left 4x4 stays zero
    }
    tile[linear] = val * invs;
  }

  const int half = lane >> 4;       // 0: lanes 0-15, 1: lanes 16-31
  const int l    = lane & 15;

  // A fragments of X (ISA A-layout: lane = row, VGPR0/1 = K {0,1} or {2,3})
  v2f A[4];
#pragma unroll
  for (int q = 0; q < 4; ++q) {
    A[q].x = tile[l * 16 + 4*q + 2*half + 0];
    A[q].y = tile[l * 16 + 4*q + 2*half + 1];
  }

  // Identity contribution in C/D layout: element (m = v + 8*half, n = l)
  float dg[8];
#pragma unroll
  for (int v = 0; v < 8; ++v) dg[v] = ((v + 8*half) == l) ? 1.0f : 0.0f;

  // Q = I + X / NTERMS  (C/D layout in registers)
  v8f Q;
#pragma unroll
  for (int v = 0; v < 8; ++v)
    Q[v] = tile[(v + 8*half) * 16 + l] * (1.0f / (float)NTERMS) + dg[v];

  // Horner Taylor: Q = I + (X @ Q)/k, k = NTERMS-1 .. 1
  for (int k = NTERMS - 1; k >= 1; --k) {
#pragma unroll
    for (int v = 0; v < 8; ++v) tile[(v + 8*half) * 16 + l] = Q[v];
    const v8f acc = mm16(A, tile, half, l);
    const float invk = 1.0f / (float)k;
#pragma unroll
    for (int v = 0; v < 8; ++v) Q[v] = acc[v] * invk + dg[v];
  }

  // Squaring phase: Q = Q @ Q, NSQUARE times
  for (int t = 0; t < NSQUARE; ++t) {
#pragma unroll
    for (int v = 0; v < 8; ++v) tile[(v + 8*half) * 16 + l] = Q[v];
    v2f Aq[4];
#pragma unroll
    for (int q = 0; q < 4; ++q) {
      Aq[q].x = tile[l * 16 + 4*q + 2*half + 0];
      Aq[q].y = tile[l * 16 + 4*q + 2*half + 1];
    }
    Q = mm16(Aq, tile, half, l);
  }

  // Publish expm(M) tile (block-diagonal: [0:8,0:8]=P0, [8:16,8:16]=P1)
#pragma unroll
  for (int v = 0; v < 8; ++v) tile[(v + 8*half) * 16 + l] = Q[v];

  // Tail: lane 0 -> problem 0, lane 1 -> problem 1. Pure scalar 4x4 work.
  if (lane < 2 && (lane == 0 || has1)) {
    const int jj = lane ? j1 : j0;
    const int ii = lane ? i1 : i0;
    const int base = lane * 8;

    float dR[4][4];
#pragma unroll
    for (int r = 0; r < 4; ++r)
#pragma unroll
      for (int c = 0; c < 4; ++c)
        dR[r][c] = tile[(base + r) * 16 + base + 4 + c];

    const size_t st = (size_t)(4 * ii + boff);
    float Rb[4][4], Cb[4][4];
#pragma unroll
    for (int r = 0; r < 4; ++r)
#pragma unroll
      for (int c = 0; c < 4; ++c) {
        Rb[r][c] = upd[(st + r) * (size_t)SROW + st + c];
        Cb[r][c] = cov[(st + r) * (size_t)SROW + st + c];
      }

    float F[4][4];                       // F = dR @ R^T
#pragma unroll
    for (int r = 0; r < 4; ++r)
#pragma unroll
      for (int c = 0; c < 4; ++c)
        F[r][c] = dR[r][0]*Rb[c][0] + dR[r][1]*Rb[c][1]
                + dR[r][2]*Rb[c][2] + dR[r][3]*Rb[c][3];

    float comm10 = 0.0f, comm32 = 0.0f;  // (F C - C F)[1,0], [3,2]
#pragma unroll
    for (int k = 0; k < 4; ++k) {
      comm10 += F[1][k]*Cb[k][0] - Cb[1][k]*F[k][0];
      comm32 += F[3][k]*Cb[k][2] - Cb[3][k]*F[k][2];
    }
    const float pe = pcpa[2 * ii], po = pcpa[2 * ii + 1];
    out[jj * ng + ii] = comm10 * pe + comm32 * po;
  }
}

extern "C" void kernel_launch(void* const* d_in, const int* in_sizes, int n_in,
                              void* d_out, int out_size, void* d_ws, size_t ws_size,
                              hipStream_t stream) {
  (void)in_sizes; (void)n_in; (void)d_ws; (void)ws_size; (void)out_size;
  const float* pcpa  = (const float*)d_in[0];
  const float* cov   = (const float*)d_in[1];
  const float* upd   = (const float*)d_in[2];
  const float* chi   = (const float*)d_in[3];
  const int*   layer = (const int*)d_in[6];
  float* out = (float*)d_out;

  const int max_pairs = (6 * (NQ / 2) + 1) / 2;                    // 6144 waves
  const int blocks = (max_pairs + WAVES_PER_BLOCK - 1) / WAVES_PER_BLOCK;  // 1536
  brickwall_expm_kernel<<<blocks, WAVES_PER_BLOCK * 32, 0, stream>>>(
      pcpa, cov, upd, chi, layer, out);
}